// RNN_75153337745427
// MI455X (gfx1250) — compile-verified
//
#include <hip/hip_runtime.h>

typedef __attribute__((ext_vector_type(2))) float v2f;
typedef __attribute__((ext_vector_type(4))) float v4f;
typedef __attribute__((ext_vector_type(8))) float v8f;
typedef int v4i_gcc __attribute__((vector_size(16)));
typedef __attribute__((address_space(1))) v4i_gcc* as1_v4i;
typedef __attribute__((address_space(3))) v4i_gcc* as3_v4i;

#define B_ 32
#define T_ 2048
#define I_ 512
#define H_ 512
#define M_TOTAL (B_ * T_) /* 65536 */
#define NWG2 16           /* workgroups in scan phase */
#define NSLICE (H_ / NWG2)
#define HP 516            /* LDS row pitch for h: 16B-aligned rows, bank-spread */

#if defined(__has_builtin)
#if __has_builtin(__builtin_amdgcn_global_load_async_to_lds_b128) && \
    __has_builtin(__builtin_amdgcn_s_wait_asynccnt)
#define ASYNC_OK 1
#endif
#endif

// ---------------- Phase 0: init state + barrier ----------------
__global__ void rnn_init(const float* __restrict__ h0,
                         float* __restrict__ hbuf,
                         unsigned* __restrict__ bar) {
    int idx = blockIdx.x * blockDim.x + threadIdx.x;
    if (idx == 0) *bar = 0u;
    if (idx < B_ * H_) hbuf[idx] = h0[idx];
}

// ---------------- Phase 1: xw = x @ W_ih^T  (into d_out) ----------------
// Grid: (M_TOTAL/128, H_/64), 256 threads (8 waves). Each wave: 16(M) x 64(N).
__global__ __launch_bounds__(256)
void rnn_proj(const float* __restrict__ x,
              const float* __restrict__ Wih,
              float* __restrict__ out) {
    const int tid  = threadIdx.x;
    const int lane = tid & 31;
    const int wave = tid >> 5;
    const int m0   = blockIdx.x * 128 + wave * 16;
    const int n0   = blockIdx.y * 64;
    const int half = lane >> 4;   // 0 or 1
    const int lrow = lane & 15;
    const int koff = half * 2;

    const float* arow = x + (size_t)(m0 + lrow) * I_;
    const float* b0 = Wih + (size_t)(n0 +  0 + lrow) * I_;
    const float* b1 = Wih + (size_t)(n0 + 16 + lrow) * I_;
    const float* b2 = Wih + (size_t)(n0 + 32 + lrow) * I_;
    const float* b3 = Wih + (size_t)(n0 + 48 + lrow) * I_;

    v8f acc0 = {}, acc1 = {}, acc2 = {}, acc3 = {};

    #pragma unroll 4
    for (int k0 = 0; k0 < I_; k0 += 4) {
        v2f a  = *(const v2f*)(arow + k0 + koff);
        v2f v0 = *(const v2f*)(b0   + k0 + koff);
        v2f v1 = *(const v2f*)(b1   + k0 + koff);
        v2f v2 = *(const v2f*)(b2   + k0 + koff);
        v2f v3 = *(const v2f*)(b3   + k0 + koff);
        acc0 = __builtin_amdgcn_wmma_f32_16x16x4_f32(false, a, false, v0, (short)0, acc0, false, false);
        acc1 = __builtin_amdgcn_wmma_f32_16x16x4_f32(false, a, false, v1, (short)0, acc1, false, false);
        acc2 = __builtin_amdgcn_wmma_f32_16x16x4_f32(false, a, false, v2, (short)0, acc2, false, false);
        acc3 = __builtin_amdgcn_wmma_f32_16x16x4_f32(false, a, false, v3, (short)0, acc3, false, false);
    }

    // C layout: VGPR r -> M = r (+8 for upper lane half), N = lane%16.
    #pragma unroll
    for (int r = 0; r < 8; ++r) {
        int m = m0 + r + half * 8;
        float* orow = out + (size_t)m * H_ + n0 + lrow;
        orow[0]  = acc0[r];
        orow[16] = acc1[r];
        orow[32] = acc2[r];
        orow[48] = acc3[r];
    }
}

// ---------------- Phase 2: recurrent scan (persistent, 16 WGs) ----------------
// Each WG owns 32 columns of H. 8 waves: 4 tiles (2 M x 2 N) x 2-way K split.
// h (32x512 f32) is staged into LDS each step via async copy; A-fragments are
// fed from LDS, B-fragments (W_hh) from L2-resident global memory.
__global__ __launch_bounds__(256)
void rnn_scan(const float* __restrict__ Whh,
              float* __restrict__ out,
              float* __restrict__ hbuf,
              unsigned* __restrict__ bar) {
    __shared__ float hs[B_ * HP];     // staged h, padded rows (64.5 KB)
    __shared__ float red[4][32][8];   // partial C tiles from kh==1 waves

    const int tid  = threadIdx.x;
    const int lane = tid & 31;
    const int wave = tid >> 5;
    const int tile = wave >> 1;       // 0..3
    const int kh   = wave & 1;        // K-split half
    const int mt   = tile >> 1;       // 0..1 (M tile)
    const int nt   = tile & 1;        // 0..1 (N tile)
    const int half = lane >> 4;
    const int lrow = lane & 15;
    const int n0   = blockIdx.x * NSLICE;

    const int m_a   = mt * 16 + lrow;           // A-fragment row (batch)
    const int n_g   = n0 + nt * 16 + lrow;      // global output column
    const int kbase = kh * (H_ / 2);
    const int koff  = half * 2;
    // B[k][n] = Whh[n][k] -> a lane's two B values are contiguous in Whh row n_g
    const float* brow = Whh + (size_t)n_g * H_ + kbase;
    const float* ash  = hs + m_a * HP + kbase;  // LDS A row for this lane

    for (int t = 0; t < T_; ++t) {
        const float* h  = hbuf + (size_t)(t & 1) * (B_ * H_);
        float*       hn = hbuf + (size_t)((t + 1) & 1) * (B_ * H_);

        // ---- stage h: global -> LDS (async copy path on CDNA5) ----
        #pragma unroll
        for (int i = tid; i < B_ * (H_ / 4); i += 256) {
            int row = i >> 7;               // 0..31
            int c4  = (i & 127) << 2;       // float col, multiple of 4 (16B)
#ifdef ASYNC_OK
            __builtin_amdgcn_global_load_async_to_lds_b128(
                (as1_v4i)(v4i_gcc*)(h + row * H_ + c4),
                (as3_v4i)(v4i_gcc*)(hs + row * HP + c4),
                0, 0);
#else
            *(v4f*)(hs + row * HP + c4) = *(const v4f*)(h + row * H_ + c4);
#endif
        }
#ifdef ASYNC_OK
        __builtin_amdgcn_s_wait_asynccnt(0);
#endif
        __syncthreads();

        // ---- GEMM tile: two independent accumulator chains ----
        v8f accA = {}, accB = {};
        #pragma unroll 8
        for (int k0 = 0; k0 < H_ / 2; k0 += 8) {
            v2f a0 = *(const v2f*)(ash  + k0 + koff);
            v2f b0 = *(const v2f*)(brow + k0 + koff);
            accA = __builtin_amdgcn_wmma_f32_16x16x4_f32(false, a0, false, b0, (short)0, accA, false, false);
            v2f a1 = *(const v2f*)(ash  + k0 + 4 + koff);
            v2f b1 = *(const v2f*)(brow + k0 + 4 + koff);
            accB = __builtin_amdgcn_wmma_f32_16x16x4_f32(false, a1, false, b1, (short)0, accB, false, false);
        }
        v8f acc = accA + accB;

        if (kh == 1) {
            #pragma unroll
            for (int r = 0; r < 8; ++r) red[tile][lane][r] = acc[r];
        }
        __syncthreads();

        if (kh == 0) {
            #pragma unroll
            for (int r = 0; r < 8; ++r) {
                int m = mt * 16 + r + half * 8;     // batch index
                int n = n0 + nt * 16 + lrow;        // hidden index
                size_t oidx = (size_t)m * (T_ * H_) + (size_t)t * H_ + n;
                float v = acc[r] + red[tile][lane][r] + out[oidx];
                v = fmaxf(v, 0.0f);
                out[oidx] = v;                      // output[b][t][h] = h_t
                hn[m * H_ + n] = v;                 // next-step state
                if (t == T_ - 1)
                    out[(size_t)B_ * T_ * H_ + (size_t)m * H_ + n] = v;  // h_n
            }
        }

        // -------- grid-wide barrier across the 16 persistent WGs --------
        __threadfence();      // release our h stores to device scope
        __syncthreads();      // also protects `red`/`hs` reuse next iteration
        if (tid == 0) {
            __hip_atomic_fetch_add(bar, 1u, __ATOMIC_RELEASE, __HIP_MEMORY_SCOPE_AGENT);
            const unsigned target = (unsigned)(t + 1) * NWG2;
            while (__hip_atomic_load(bar, __ATOMIC_ACQUIRE, __HIP_MEMORY_SCOPE_AGENT) < target)
                __builtin_amdgcn_s_sleep(2);
        }
        __syncthreads();
        __threadfence();      // acquire: invalidate near caches before reading peers' h
    }
}

// ---------------- Host launcher ----------------
extern "C" void kernel_launch(void* const* d_in, const int* in_sizes, int n_in,
                              void* d_out, int out_size, void* d_ws, size_t ws_size,
                              hipStream_t stream) {
    const float* x   = (const float*)d_in[0];   // (B,T,I)
    const float* h0  = (const float*)d_in[1];   // (1,B,H)
    const float* Wih = (const float*)d_in[2];   // (H,I)
    const float* Whh = (const float*)d_in[3];   // (H,H)
    float* out = (float*)d_out;                 // (B,T,H) then (1,B,H)

    float*    hbuf = (float*)d_ws;                                   // 2 x B x H
    unsigned* bar  = (unsigned*)((char*)d_ws + 2u * B_ * H_ * sizeof(float));

    rnn_init<<<(B_ * H_ + 255) / 256, 256, 0, stream>>>(h0, hbuf, bar);

    dim3 g1(M_TOTAL / 128, H_ / 64);
    rnn_proj<<<g1, 256, 0, stream>>>(x, Wih, out);

    rnn_scan<<<NWG2, 256, 0, stream>>>(Whh, out, hbuf, bar);
}